// Dynamic_conv1d_85487029059636
// MI455X (gfx1250) — compile-verified
//
#include <hip/hip_runtime.h>
#include <math.h>

// ---------------- problem constants ----------------
#define B_SZ   4096
#define L_IN   64
#define OUT_C  64
#define KS_SZ  8
#define K_SZ   4
#define L_OUT  57
#define C_CH   3648          // OUT_C * L_OUT
#define HID    33
#define TEMP_INV (1.0f/34.0f)
#define EPS_BN 1e-5f

// d_out layout: [ out (B*C) | attn (B*4) | kw_t (B*4*C) ]
#define OFF_ATTN ((size_t)B_SZ * C_CH)
#define OFF_KW   (OFF_ATTN + (size_t)B_SZ * K_SZ)

// tiling
#define NB_STAT 64                      // b-tiles in stats pass
#define BB_STAT (B_SZ / NB_STAT)        // 64 rows per stats wave
#define NB_OUT  256                     // b-tiles in output pass
#define BB_OUT  (B_SZ / NB_OUT)         // 16 rows per output wave
#define NBC ((size_t)NB_STAT * C_CH)    // one partial-array stride

typedef float v8f __attribute__((ext_vector_type(8)));
typedef float v2f __attribute__((ext_vector_type(2)));

__device__ __forceinline__ float gelu_f(float x) {
  // exact gelu (approximate=False): 0.5*x*(1+erf(x/sqrt(2)))
  return 0.5f * x * (1.0f + erff(x * 0.70710678118654752440f));
}

__device__ __forceinline__ v8f wmma_f32_16x16x4(v2f a, v2f b, v8f c) {
  // V_WMMA_F32_16X16X4_F32: D(16x16 f32) = A(16x4) * B(4x16) + C
  return __builtin_amdgcn_wmma_f32_16x16x4_f32(false, a, false, b, (short)0, c,
                                               false, false);
}

__device__ __forceinline__ float sum4(float v) {
  v += __shfl_xor(v, 1, 32);
  v += __shfl_xor(v, 2, 32);
  return v;
}

// ---------------------------------------------------------------
// Kernel A: attn = softmax( gelu(x @ fc1_w^T) @ fc2_w^T + fc2_b , /TEMP )
// one thread per batch row
// ---------------------------------------------------------------
__global__ __launch_bounds__(256) void attn_kernel(
    const float* __restrict__ x, const float* __restrict__ fc1_w,
    const float* __restrict__ fc2_w, const float* __restrict__ fc2_b,
    float* __restrict__ attn_out) {
  __shared__ float w1[HID * L_IN];   // 33x64
  __shared__ float w2[K_SZ * HID];   // 4x33
  __shared__ float b2[K_SZ];
  const int t = threadIdx.x;
  for (int i = t; i < HID * L_IN; i += 256) w1[i] = fc1_w[i];
  for (int i = t; i < K_SZ * HID; i += 256) w2[i] = fc2_w[i];
  if (t < K_SZ) b2[t] = fc2_b[t];
  __syncthreads();

  const int b = blockIdx.x * 256 + t;
  const float* xr = x + (size_t)b * L_IN;

  float a[HID];
#pragma unroll
  for (int h = 0; h < HID; ++h) a[h] = 0.0f;
  for (int j = 0; j < L_IN; ++j) {
    const float xv = xr[j];
#pragma unroll
    for (int h = 0; h < HID; ++h) a[h] = fmaf(xv, w1[h * L_IN + j], a[h]);
  }
#pragma unroll
  for (int h = 0; h < HID; ++h) a[h] = gelu_f(a[h]);

  float lg[K_SZ];
#pragma unroll
  for (int k = 0; k < K_SZ; ++k) {
    float s = b2[k];
#pragma unroll
    for (int h = 0; h < HID; ++h) s = fmaf(a[h], w2[k * HID + h], s);
    lg[k] = s * TEMP_INV;
  }
  const float mx = fmaxf(fmaxf(lg[0], lg[1]), fmaxf(lg[2], lg[3]));
  float e[K_SZ], den = 0.0f;
#pragma unroll
  for (int k = 0; k < K_SZ; ++k) { e[k] = __expf(lg[k] - mx); den += e[k]; }
  const float r = 1.0f / den;
#pragma unroll
  for (int k = 0; k < K_SZ; ++k) attn_out[(size_t)b * K_SZ + k] = e[k] * r;
}

// ---------------------------------------------------------------
// Kernel B (pass 1, WMMA): per-channel partial sums for both batchnorms.
// grid.x = 16 o-tiles (4 o each -> 16 (o,k) columns), grid.y = NB_STAT b-tiles.
// one wave per block. Deterministic: each (btile,channel) partial written once.
// ---------------------------------------------------------------
__global__ __launch_bounds__(32) void stats_kernel(
    const float* __restrict__ x, const float* __restrict__ weight,
    const float* __restrict__ bias, const float* __restrict__ attn,
    float* __restrict__ part) {
  const int ot = blockIdx.x;          // 0..15
  const int bt = blockIdx.y;          // 0..NB_STAT-1
  const int o0 = ot * 4;
  const int t = threadIdx.x;
  const int tlo = t & 15;
  const int hi = (t >= 16) ? 1 : 0;
  const int koff = hi * 2;

  __shared__ float wlds[8 * 16];      // [c][n], n = o_local*4 + k
  __shared__ float xl[72];            // one x row, zero-padded

  for (int idx = t; idx < 128; idx += 32) {
    const int c = idx >> 4, n = idx & 15;
    const int k = n & 3, o = o0 + (n >> 2);
    wlds[idx] = weight[((size_t)(k * OUT_C + o)) * KS_SZ + c];
  }
  __syncthreads();

  // loop-invariant B operands (weight slice) and bias for this lane's column
  const v2f bw0 = {wlds[(0 + koff) * 16 + tlo], wlds[(1 + koff) * 16 + tlo]};
  const v2f bw1 = {wlds[(4 + koff) * 16 + tlo], wlds[(5 + koff) * 16 + tlo]};
  const int kk = tlo & 3;
  const int ol = tlo >> 2;
  const float biasv = bias[kk * OUT_C + o0 + ol];

  const int b0 = bt * BB_STAT;

  for (int l0p = 0; l0p < 4; ++l0p) {
    const int l0 = l0p * 16;
    float ks[8], kq[8], os[8], oq[8];
#pragma unroll
    for (int i = 0; i < 8; ++i) { ks[i] = kq[i] = os[i] = oq[i] = 0.0f; }

    for (int bb = 0; bb < BB_STAT; ++bb) {
      const int b = b0 + bb;
      xl[t] = x[(size_t)b * L_IN + t];
      xl[t + 32] = x[(size_t)b * L_IN + 32 + t];
      if (t < 8) xl[64 + t] = 0.0f;
      __syncthreads();

      const int abase = l0 + tlo + koff;
      const v2f a0 = {xl[abase + 0], xl[abase + 1]};
      const v2f a1 = {xl[abase + 4], xl[abase + 5]};
      v8f acc;
#pragma unroll
      for (int i = 0; i < 8; ++i) acc[i] = biasv;     // C preloaded with bias
      acc = wmma_f32_16x16x4(a0, bw0, acc);           // c = 0..3
      acc = wmma_f32_16x16x4(a1, bw1, acc);           // c = 4..7

      const float aw = attn[(size_t)b * K_SZ + kk];
#pragma unroll
      for (int i = 0; i < 8; ++i) {
        const int l = l0 + i + hi * 8;
        const float e = acc[i];                       // kw pre-act (w/ bias)
        const float s = sum4(aw * e);                 // out pre-act (sum over k)
        if (l < L_OUT) {
          const float g2 = gelu_f(e);
          ks[i] += g2; kq[i] += g2 * g2;
          const float g = gelu_f(s);
          os[i] += g; oq[i] += g * g;                 // identical in k-group
        }
      }
      __syncthreads();
    }

#pragma unroll
    for (int i = 0; i < 8; ++i) {
      const float kS = sum4(ks[i]);                   // fold k into channel
      const float kQ = sum4(kq[i]);
      const int l = l0 + i + hi * 8;
      if ((tlo & 3) == 0 && l < L_OUT) {
        const int ch = (o0 + ol) * L_OUT + l;
        const size_t base = (size_t)bt * C_CH + ch;
        part[base]           = kS;
        part[NBC + base]     = kQ;
        part[2 * NBC + base] = os[i];
        part[3 * NBC + base] = oq[i];
      }
    }
  }
}

// ---------------------------------------------------------------
// Kernel C: fold partials -> per-channel scale/shift for both BNs
// ---------------------------------------------------------------
__global__ __launch_bounds__(256) void finalize_kernel(
    const float* __restrict__ part, const float* __restrict__ gamma,
    const float* __restrict__ beta, float* __restrict__ sc) {
  const int c = blockIdx.x * 256 + threadIdx.x;
  if (c >= C_CH) return;
  float ks = 0, kq = 0, os = 0, oq = 0;
  for (int bt = 0; bt < NB_STAT; ++bt) {             // fixed order: deterministic
    const size_t base = (size_t)bt * C_CH + c;
    ks += part[base];
    kq += part[NBC + base];
    os += part[2 * NBC + base];
    oq += part[3 * NBC + base];
  }
  const float g = gamma[c], bt0 = beta[c];
  const float invK = 1.0f / (float)(B_SZ * K_SZ);    // kw BN over (B,K)
  const float mk = ks * invK;
  const float vk = kq * invK - mk * mk;
  const float sck = rsqrtf(vk + EPS_BN) * g;
  const float invB = 1.0f / (float)B_SZ;             // out BN over B
  const float mo = os * invB;
  const float vo = oq * invB - mo * mo;
  const float sco = rsqrtf(vo + EPS_BN) * g;
  sc[c]             = sck;
  sc[C_CH + c]      = bt0 - mk * sck;
  sc[2 * C_CH + c]  = sco;
  sc[3 * C_CH + c]  = bt0 - mo * sco;
}

// ---------------------------------------------------------------
// Kernel D (pass 2, WMMA): recompute tiles, normalize, and write
// `out` (B,C) and `kw_t` (B,K,C) with LDS-staged coalesced stores.
// ---------------------------------------------------------------
__global__ __launch_bounds__(32) void out_kernel(
    const float* __restrict__ x, const float* __restrict__ weight,
    const float* __restrict__ bias, const float* __restrict__ attn,
    const float* __restrict__ sc, float* __restrict__ dst) {
  const int ot = blockIdx.x;
  const int bt = blockIdx.y;
  const int o0 = ot * 4;
  const int t = threadIdx.x;
  const int tlo = t & 15;
  const int hi = (t >= 16) ? 1 : 0;
  const int koff = hi * 2;

  __shared__ float wlds[8 * 16];
  __shared__ float xl[72];
  __shared__ float sck[228], shk[228], sco[228], sho[228];  // local channels
  __shared__ float kw_tile[K_SZ * 228];                     // [k][o_l*57+l]
  __shared__ float out_tile[228];

  for (int idx = t; idx < 128; idx += 32) {
    const int c = idx >> 4, n = idx & 15;
    const int k = n & 3, o = o0 + (n >> 2);
    wlds[idx] = weight[((size_t)(k * OUT_C + o)) * KS_SZ + c];
  }
  const int cbase = ot * 228;                  // channels are contiguous per o-tile
  for (int idx = t; idx < 228; idx += 32) {
    sck[idx] = sc[cbase + idx];
    shk[idx] = sc[C_CH + cbase + idx];
    sco[idx] = sc[2 * C_CH + cbase + idx];
    sho[idx] = sc[3 * C_CH + cbase + idx];
  }
  __syncthreads();

  const v2f bw0 = {wlds[(0 + koff) * 16 + tlo], wlds[(1 + koff) * 16 + tlo]};
  const v2f bw1 = {wlds[(4 + koff) * 16 + tlo], wlds[(5 + koff) * 16 + tlo]};
  const int kk = tlo & 3;
  const int ol = tlo >> 2;
  const float biasv = bias[kk * OUT_C + o0 + ol];

  float* __restrict__ outp = dst;
  float* __restrict__ kwp = dst + OFF_KW;
  const int b0 = bt * BB_OUT;

  for (int bb = 0; bb < BB_OUT; ++bb) {
    const int b = b0 + bb;
    xl[t] = x[(size_t)b * L_IN + t];
    xl[t + 32] = x[(size_t)b * L_IN + 32 + t];
    if (t < 8) xl[64 + t] = 0.0f;
    __syncthreads();

    const float aw = attn[(size_t)b * K_SZ + kk];

    for (int l0p = 0; l0p < 4; ++l0p) {
      const int l0 = l0p * 16;
      const int abase = l0 + tlo + koff;
      const v2f a0 = {xl[abase + 0], xl[abase + 1]};
      const v2f a1 = {xl[abase + 4], xl[abase + 5]};
      v8f acc;
#pragma unroll
      for (int i = 0; i < 8; ++i) acc[i] = biasv;
      acc = wmma_f32_16x16x4(a0, bw0, acc);
      acc = wmma_f32_16x16x4(a1, bw1, acc);

#pragma unroll
      for (int i = 0; i < 8; ++i) {
        const int l = l0 + i + hi * 8;
        const float e = acc[i];
        const float s = sum4(aw * e);
        if (l < L_OUT) {
          const int j = ol * L_OUT + l;              // local channel
          kw_tile[kk * 228 + j] = gelu_f(e) * sck[j] + shk[j];
          if (kk == 0) out_tile[j] = gelu_f(s) * sco[j] + sho[j];
        }
      }
    }
    __syncthreads();

    // coalesced stores: per (b,k) the 228 local channels are contiguous
    for (int idx = t; idx < K_SZ * 228; idx += 32) {
      const int k = idx / 228;
      const int rem = idx - k * 228;
      kwp[((size_t)(b * K_SZ + k)) * C_CH + cbase + rem] = kw_tile[idx];
    }
    for (int idx = t; idx < 228; idx += 32) {
      outp[(size_t)b * C_CH + cbase + idx] = out_tile[idx];
    }
    __syncthreads();
  }
}

// ---------------------------------------------------------------
extern "C" void kernel_launch(void* const* d_in, const int* in_sizes, int n_in,
                              void* d_out, int out_size, void* d_ws,
                              size_t ws_size, hipStream_t stream) {
  const float* x      = (const float*)d_in[0];
  const float* fc1_w  = (const float*)d_in[1];
  const float* fc2_w  = (const float*)d_in[2];
  const float* fc2_b  = (const float*)d_in[3];
  const float* weight = (const float*)d_in[4];
  const float* bias   = (const float*)d_in[5];
  const float* gamma  = (const float*)d_in[6];
  const float* beta   = (const float*)d_in[7];

  float* out  = (float*)d_out;
  float* attn = out + OFF_ATTN;

  float* part   = (float*)d_ws;            // 4 * NB_STAT * C_CH floats
  float* scales = part + 4 * NBC;          // 4 * C_CH floats

  attn_kernel<<<B_SZ / 256, 256, 0, stream>>>(x, fc1_w, fc2_w, fc2_b, attn);
  stats_kernel<<<dim3(16, NB_STAT), 32, 0, stream>>>(x, weight, bias, attn,
                                                     part);
  finalize_kernel<<<(C_CH + 255) / 256, 256, 0, stream>>>(part, gamma, beta,
                                                          scales);
  out_kernel<<<dim3(16, NB_OUT), 32, 0, stream>>>(x, weight, bias, attn,
                                                  scales, out);
}